// DecoderTransformer_3925600108956
// MI455X (gfx1250) — compile-verified
//
#include <hip/hip_runtime.h>
#include <hip/hip_bf16.h>
#include <stdint.h>

// Problem constants (from the reference)
namespace {
constexpr int       kB          = 8;
constexpr int       kS          = 2048;
constexpr int       kN          = 512;
constexpr int       kK          = 4;
constexpr int       kDG         = 512;
constexpr int       kDSEQ       = 512;
constexpr int       kDOUT       = kDG + kDSEQ;            // 1024
constexpr int       kRows       = kB * kS;                // 16384 (b*s rows)
constexpr long long kEncElems   = (long long)kRows * kDOUT; // 16,777,216 floats
constexpr int       kRowF4      = kDOUT / 4;              // 256 float4 per out row
constexpr int       kGraphF4Tot = kRows * (kDG / 4);      // 2,097,152
constexpr int       kSeqF4Tot   = kRows * (kDSEQ / 4);    // 2,097,152
constexpr int       kHidF4      = (kB * 1024) / 4;        // 2048
constexpr float     kEps        = 1e-8f;
constexpr int       kCopyPerThr = 4;                      // float4s per thread in copy
}

// ---------------- CDNA5 async-LDS helpers (device-compile only) -------------
#if defined(__HIP_DEVICE_COMPILE__)
typedef int v4i __attribute__((ext_vector_type(4)));
typedef __attribute__((address_space(1))) void* as1_void;
typedef __attribute__((address_space(3))) void* as3_void;
typedef __attribute__((address_space(1))) v4i*  as1_v4i;
typedef __attribute__((address_space(3))) v4i*  as3_v4i;

__device__ __forceinline__ void async_wait0() {
#if __has_builtin(__builtin_amdgcn_s_wait_asynccnt)
  __builtin_amdgcn_s_wait_asynccnt(0);
#else
  asm volatile("s_wait_asynccnt 0" ::: "memory");
#endif
}

// global memory -> LDS, 128-bit, ASYNCcnt-tracked
__device__ __forceinline__ void async_g2l_b128(const void* g, void* l) {
#if __has_builtin(__builtin_amdgcn_global_load_async_to_lds_b128)
  __builtin_amdgcn_global_load_async_to_lds_b128(
      (as1_v4i)(as1_void)(void*)g, (as3_v4i)(as3_void)l, 0, 0);
#else
  unsigned lds_off = (unsigned)(uintptr_t)(as3_void)l;
  asm volatile("global_load_async_to_lds_b128 %0, %1, off"
               :: "v"(lds_off), "v"(g) : "memory");
#endif
}

// LDS -> global memory, 128-bit, ASYNCcnt-tracked
__device__ __forceinline__ void async_l2g_b128(void* g, const void* l) {
#if __has_builtin(__builtin_amdgcn_global_store_async_from_lds_b128)
  __builtin_amdgcn_global_store_async_from_lds_b128(
      (as1_v4i)(as1_void)g, (as3_v4i)(as3_void)(void*)l, 0, 0);
#else
  unsigned lds_off = (unsigned)(uintptr_t)(as3_void)(void*)l;
  asm volatile("global_store_async_from_lds_b128 %0, %1, off"
               :: "v"(g), "v"(lds_off) : "memory");
#endif
}
#endif  // __HIP_DEVICE_COMPILE__

// ---------------------------------------------------------------------------
// Kernel (listed first so the disasm snippet shows the async path):
// bulk 32MB seq_output copy into the upper half of each out row, staged
// through LDS with the gfx1250 async-LDS instructions. Each 256-thread block
// moves a 16KB tile: 4 async loads/thread -> one s_wait_asynccnt -> 4 async
// stores/thread. Deep ASYNCcnt queue, no VGPR round trip.
__global__ void __launch_bounds__(256) k_copy_seq(const float4* __restrict__ seq4,
                                                  float* __restrict__ out) {
  __shared__ float4 stage[256 * kCopyPerThr];   // 16 KB
  const int t    = threadIdx.x;
  const int base = blockIdx.x * (256 * kCopyPerThr);
#if defined(__HIP_DEVICE_COMPILE__)
#pragma unroll
  for (int u = 0; u < kCopyPerThr; ++u) {
    int gid = base + u * 256 + t;               // [0, kSeqF4Tot)
    async_g2l_b128(seq4 + gid, &stage[u * 256 + t]);     // ASYNCcnt++
  }
  async_wait0();                                 // all 4 loads landed in LDS
#pragma unroll
  for (int u = 0; u < kCopyPerThr; ++u) {
    int gid = base + u * 256 + t;
    int row = gid >> 7;                          // / (kDSEQ/4)
    int col = gid & 127;
    float4* dst = (float4*)(out + (long long)row * kDOUT + kDSEQ) + col;
    async_l2g_b128(dst, &stage[u * 256 + t]);    // ASYNCcnt++
  }
  async_wait0();                                 // (S_ENDPGM also waits idle)
#else
  for (int u = 0; u < kCopyPerThr; ++u) {        // host-pass placeholder
    int gid = base + u * 256 + t;
    int row = gid >> 7;
    int col = gid & 127;
    *((float4*)(out + (long long)row * kDOUT + kDSEQ) + col) = seq4[gid];
  }
#endif
}

// Kernel: scatter-add. One block per (b, n). 256 threads * float2 = 512 f32.
// Hardware f32 atomics into the graph half of d_out; int atomics for counts.
__global__ void __launch_bounds__(256) k_scatter(const float2* __restrict__ graph2,
                                                 const int* __restrict__ idx,
                                                 float* __restrict__ out,
                                                 unsigned* __restrict__ counts) {
  int bn = blockIdx.x;        // [0, B*N)
  int b  = bn >> 9;           // / N
  int t  = threadIdx.x;       // [0, 256)
  __shared__ int sidx[kK];
  if (t < kK) sidx[t] = idx[bn * kK + t];
  float2 g = graph2[(long long)bn * (kDG / 2) + t];
  if (bn + 1 < kB * kN)  // gfx1250 global_prefetch_b8 for the next row
    __builtin_prefetch(graph2 + (long long)(bn + 1) * (kDG / 2) + t, 0, 0);
  __syncthreads();
#pragma unroll
  for (int k = 0; k < kK; ++k) {
    long long srow = (long long)b * kS + sidx[k];
    float* dst = out + srow * kDOUT + t * 2;
    unsafeAtomicAdd(dst,     g.x);   // global_atomic_add_f32
    unsafeAtomicAdd(dst + 1, g.y);
    if (t == 0) atomicAdd(&counts[srow], 1u);
  }
}

// Kernel: zero the graph half of every output row + zero the counts in ws.
// (Must run every call: harness does not re-poison between graph replays.)
__global__ void __launch_bounds__(256) k_zero(float4* __restrict__ out4,
                                              unsigned* __restrict__ counts) {
  int gid = blockIdx.x * 256 + threadIdx.x;   // [0, kGraphF4Tot)
  int row = gid >> 7;                         // / (kDG/4)
  int col = gid & 127;
  out4[(long long)row * kRowF4 + col] = make_float4(0.f, 0.f, 0.f, 0.f);
  if (gid < kRows) counts[gid] = 0u;
}

// Kernel: graph_hidden = (sums + eps) / max(count, 1). One block per row.
__global__ void __launch_bounds__(128) k_finalize(float* __restrict__ out,
                                                  const unsigned* __restrict__ counts) {
  int row = blockIdx.x;
  int t   = threadIdx.x;  // [0,128): one float4 of the 512-float graph half
  float inv = 1.0f / fmaxf((float)counts[row], 1.0f);
  float4* p = (float4*)(out + (long long)row * kDOUT) + t;
  float4 v = *p;
  v.x = (v.x + kEps) * inv;
  v.y = (v.y + kEps) * inv;
  v.z = (v.z + kEps) * inv;
  v.w = (v.w + kEps) * inv;
  *p = v;
}

// Kernel: hidden -> tail of d_out (verbatim reshape/copy).
__global__ void __launch_bounds__(256) k_copy_hidden(const float4* __restrict__ hid4,
                                                     float4* __restrict__ out4) {
  int gid = blockIdx.x * 256 + threadIdx.x;   // [0, kHidF4)
  out4[gid] = hid4[gid];
}

// ---------------------------------------------------------------------------
extern "C" void kernel_launch(void* const* d_in, const int* in_sizes, int n_in,
                              void* d_out, int out_size, void* d_ws, size_t ws_size,
                              hipStream_t stream) {
  (void)in_sizes; (void)n_in; (void)out_size; (void)ws_size;
  const float*  seq    = (const float*)d_in[0];   // [B,S,DSEQ]
  const float*  graph  = (const float*)d_in[1];   // [B,N,DG]
  const float*  hidden = (const float*)d_in[2];   // [B,H]
  const int*    idx    = (const int*)d_in[3];     // [B,N,K]
  float*        out    = (float*)d_out;           // enc [B,S,1024] ++ hidden [B,1024]
  unsigned*     counts = (unsigned*)d_ws;         // kRows u32 (64 KB scratch)

  k_zero       <<<kGraphF4Tot / 256,          256, 0, stream>>>((float4*)out, counts);
  k_scatter    <<<kB * kN,                    256, 0, stream>>>((const float2*)graph, idx,
                                                                out, counts);
  k_finalize   <<<kRows,                      128, 0, stream>>>(out, counts);
  k_copy_seq   <<<kSeqF4Tot / (256 * kCopyPerThr), 256, 0, stream>>>((const float4*)seq, out);
  k_copy_hidden<<<kHidF4 / 256,               256, 0, stream>>>((const float4*)hidden,
                                                                (float4*)(out + kEncElems));
}